// VectorQuantizer_32727650795873
// MI455X (gfx1250) — compile-verified
//
#include <hip/hip_runtime.h>
#include <math.h>

typedef float v2f __attribute__((ext_vector_type(2)));
typedef float v8f __attribute__((ext_vector_type(8)));

#define K_CB       512
#define C_DIM      64
#define N_TOT      131072      // 32*64*64 flattened spatial rows
#define CHUNK      64          // codebook rows per LDS chunk (8 chunks, dbl-buffered)
#define LDS_STRIDE 68          // padded row stride (floats): conflict-free b64 reads
#define OUT_ELEMS  8388608     // 32*64*64*64

// ---------------------------------------------------------------------------
// Kernel 1: half squared norms of codebook rows: cnorm[k] = 0.5 * ||c_k||^2
// ---------------------------------------------------------------------------
__global__ __launch_bounds__(256) void vq_cnorm_kernel(
    const float* __restrict__ cb, float* __restrict__ cnorm) {
  int k = blockIdx.x * 256 + threadIdx.x;
  if (k < K_CB) {
    const float* row = cb + k * C_DIM;
    float s = 0.f;
#pragma unroll
    for (int c = 0; c < C_DIM; ++c) s += row[c] * row[c];
    cnorm[k] = 0.5f * s;
  }
}

// ---------------------------------------------------------------------------
// Kernel 2: WMMA f32 GEMM (z_flat x codebook^T) with fused argmax, gather,
// quantized output in (B,C,H,W) layout, loss accumulation, and histogram.
// Block = 256 threads = 8 waves; each wave owns 16 rows (one 16x16 C tile).
// Codebook chunks are staged into LDS with GLOBAL_LOAD_ASYNC_TO_LDS_B128,
// double-buffered so chunk N+1 streams in while chunk N feeds the WMMAs.
// ---------------------------------------------------------------------------
__global__ __launch_bounds__(256) void vq_main_kernel(
    const float* __restrict__ z, const float* __restrict__ cb,
    const float* __restrict__ cnorm, float* __restrict__ out,
    float* __restrict__ loss_accum, float* __restrict__ counts) {
  __shared__ float sCB[2][CHUNK * LDS_STRIDE];  // 2 x 17.4 KB codebook chunks
  __shared__ int   sIdx[8 * 16];                // per-wave row -> codebook index

  const int tid    = threadIdx.x;
  const int lane   = tid & 31;
  const int waveId = tid >> 5;
  const int half   = lane >> 4;   // 0: lanes 0-15, 1: lanes 16-31
  const int l15    = lane & 15;

  // Row block for this wave: 16 consecutive flattened rows n = (b*64+h)*64+w.
  const int rowBase = (blockIdx.x * 8 + waveId) * 16;
  const int n       = rowBase + l15;
  const int b       = n >> 12;
  const int s       = n & 4095;            // h*64 + w
  const int zbase   = b * 262144 + s;      // + c*4096 indexes z[b,c,h,w]

  // Issue one 64-row codebook chunk into LDS buffer `buf` via async copies.
  // 64 rows x 64 f32 = 1024 float4; 256 threads -> 4 b128 async loads each.
  auto issue_chunk = [&](int chunkIdx, int buf) {
    const unsigned long long gbase =
        (unsigned long long)(uintptr_t)(cb + chunkIdx * CHUNK * C_DIM);
#pragma unroll
    for (int i = 0; i < 4; ++i) {
      int lin  = tid + i * 256;     // float4 index within chunk
      int row  = lin >> 4;
      int col4 = lin & 15;
      // Low 32 bits of the generic pointer == LDS offset (as3 aperture).
      unsigned int lds_off =
          (unsigned int)(uintptr_t)&sCB[buf][row * LDS_STRIDE + col4 * 4];
      unsigned long long gaddr = gbase + (unsigned long long)lin * 16ull;
      asm volatile("global_load_async_to_lds_b128 %0, %1, off"
                   :: "v"(lds_off), "v"(gaddr) : "memory");
    }
  };

  // ---- Load A (16x64 f32) into registers in WMMA 16x16x4_f32 A layout ----
  // A-tile kt: lanes 0-15 hold (row=l15, K=4kt+{0,1}); lanes 16-31 K=4kt+{2,3}
  v2f A[16];
#pragma unroll
  for (int kt = 0; kt < 16; ++kt) {
    int c0 = 4 * kt + 2 * half;
    A[kt].x = z[zbase + (c0 + 0) * 4096];
    A[kt].y = z[zbase + (c0 + 1) * 4096];
  }

  // Running argmax of score = z.c_k - 0.5||c_k||^2 (== argmin distance).
  float bestVal[8];
  int   bestIdx[8];
#pragma unroll
  for (int r = 0; r < 8; ++r) { bestVal[r] = -3.0e38f; bestIdx[r] = 0; }

  issue_chunk(0, 0);
  int cur = 0;
  for (int chunk = 0; chunk < 8; ++chunk) {
    asm volatile("s_wait_asynccnt 0x0" ::: "memory");  // my async fills landed
    __syncthreads();                                   // whole buffer visible
    if (chunk < 7) issue_chunk(chunk + 1, cur ^ 1);    // stream next chunk

    const float* bufp = &sCB[cur][0];
#pragma unroll
    for (int ct = 0; ct < 4; ++ct) {
      const int cb16 = chunk * CHUNK + ct * 16;
      v8f acc = {};
      // B-operand (4x16): lane reads CB[cb16 + l15][4kt + 2*half .. +1] as f32x2
      const float* bp = bufp + (ct * 16 + l15) * LDS_STRIDE + 2 * half;
#pragma unroll
      for (int kt = 0; kt < 16; ++kt) {
        v2f Bv = *(const v2f*)(bp + 4 * kt);
        acc = __builtin_amdgcn_wmma_f32_16x16x4_f32(
            /*neg_a=*/false, A[kt], /*neg_b=*/false, Bv,
            /*c_mod=*/(short)0, acc, /*reuse_a=*/false, /*reuse_b=*/false);
      }
      const int   myIdx = cb16 + l15;       // candidate column (codebook index)
      const float cn    = cnorm[myIdx];     // L2-resident
#pragma unroll
      for (int r = 0; r < 8; ++r) {
        float v = acc[r] - cn;
        if (v > bestVal[r]) { bestVal[r] = v; bestIdx[r] = myIdx; }
      }
    }
    cur ^= 1;
  }

  // ---- Cross-lane argmax within each 16-lane half (first-index tie-break) ----
#pragma unroll
  for (int r = 0; r < 8; ++r) {
    float bv = bestVal[r];
    int   bi = bestIdx[r];
#pragma unroll
    for (int off = 8; off >= 1; off >>= 1) {
      float ov = __shfl_xor(bv, off, 32);
      int   oi = __shfl_xor(bi, off, 32);
      if (ov > bv || (ov == bv && oi < bi)) { bv = ov; bi = oi; }
    }
    bestVal[r] = bv;
    bestIdx[r] = bi;
  }
  // Lanes 0 / 16 now hold winners for rows r / 8+r.
  if (l15 == 0) {
#pragma unroll
    for (int r = 0; r < 8; ++r) {
      int row = half * 8 + r;
      sIdx[waveId * 16 + row] = bestIdx[r];
      atomicAdd(&counts[bestIdx[r]], 1.0f);
    }
  }
  __syncthreads();

  // ---- Epilogue: gather codebook row, write quantized (B,C,H,W), loss ----
  const int    qIdx  = sIdx[waveId * 16 + l15];  // index for my row
  const float* cbrow = cb + qIdx * C_DIM;
  float lsum = 0.f;
#pragma unroll 8
  for (int c2 = 0; c2 < 32; ++c2) {
    int   c  = 2 * c2 + half;          // halves cover even/odd channels
    int   a  = zbase + c * 4096;       // z and out share (B,C,H,W) indexing
    float q  = cbrow[c];               // 512-row gather, L2-resident
    float zv = z[a];
    out[a]   = q;                      // straight-through == quantized value
    float d  = q - zv;
    lsum += d * d;
  }
#pragma unroll
  for (int off = 16; off >= 1; off >>= 1) lsum += __shfl_xor(lsum, off, 32);
  if (lane == 0) atomicAdd(loss_accum, lsum);
}

// ---------------------------------------------------------------------------
// Kernel 3: finalize scalars: loss = 1.25 * SSE / (N*C); perplexity.
// ---------------------------------------------------------------------------
__global__ __launch_bounds__(512) void vq_finalize_kernel(
    const float* __restrict__ loss_accum, const float* __restrict__ counts,
    float* __restrict__ out_scalars) {
  __shared__ float red[512];
  int t = threadIdx.x;
  float p = counts[t] * (1.0f / (float)N_TOT);
  red[t] = p * logf(p + 1e-10f);
  __syncthreads();
  for (int sft = 256; sft >= 1; sft >>= 1) {
    if (t < sft) red[t] += red[t + sft];
    __syncthreads();
  }
  if (t == 0) {
    out_scalars[0] = 1.25f * loss_accum[0] * (1.0f / (float)OUT_ELEMS); // loss
    out_scalars[1] = expf(-red[0]);                                    // perplexity
  }
}

// ---------------------------------------------------------------------------
extern "C" void kernel_launch(void* const* d_in, const int* in_sizes, int n_in,
                              void* d_out, int out_size, void* d_ws, size_t ws_size,
                              hipStream_t stream) {
  (void)in_sizes; (void)n_in; (void)out_size; (void)ws_size;
  const float* z  = (const float*)d_in[0];   // (32,64,64,64) f32
  const float* cb = (const float*)d_in[1];   // (512,64) f32
  float* out = (float*)d_out;                // 8388608 quantized + loss + perplexity
  float* ws  = (float*)d_ws;

  float* loss_accum = ws;             // [0]       SSE accumulator
  float* counts     = ws + 1;         // [1..512]  histogram
  float* cnorm      = ws + 1 + K_CB;  // [513..]   0.5*||c_k||^2

  hipMemsetAsync(d_ws, 0, (1 + K_CB) * sizeof(float), stream);  // loss + counts
  vq_cnorm_kernel<<<2, 256, 0, stream>>>(cb, cnorm);
  vq_main_kernel<<<N_TOT / 128, 256, 0, stream>>>(z, cb, cnorm, out,
                                                  loss_accum, counts);
  vq_finalize_kernel<<<1, 512, 0, stream>>>(loss_accum, counts, out + OUT_ELEMS);
}